// BetaBinsMassAdaptive_35296041239084
// MI455X (gfx1250) — compile-verified
//
#include <hip/hip_runtime.h>
#include <math.h>

#define NBINS 5
#define NEDGE 4   // interior edges where betainc is evaluated

// Clang extended vectors: accepted by __builtin_nontemporal_load/store
typedef int   v2i __attribute__((ext_vector_type(2)));
typedef float v2f __attribute__((ext_vector_type(2)));

__device__ __forceinline__ float frcp(float x) {
    return __builtin_amdgcn_rcpf(x);   // v_rcp_f32
}

// lgamma for x in ~[0.4, 12]: shift by 3 then Stirling with 1/(12z)-1/(360z^3).
// Abs error ~2e-6 at z=3.4, smaller above. 2 logs + ~12 FMA (vs heavyweight libm).
__device__ __forceinline__ float lgamma_pos(float x) {
    const float HALF_LN_2PI = 0.91893853320467274178f;
    float z     = x + 3.0f;
    float invz  = frcp(z);
    float invz2 = invz * invz;
    float lz    = __logf(z);
    float stir  = (z - 0.5f) * lz - z + HALF_LN_2PI
                + invz * (0.083333333333f - 0.0027777777778f * invz2);
    float p = x * (x + 1.0f) * (x + 2.0f);
    return stir - __logf(p);
}

// Regularized incomplete beta at 4 edges simultaneously.
// Branchless NR symmetry swap (bt is symmetric under (a,b,x)->(b,a,1-x)), then
// 4 independent Lentz continued fractions interleaved in ONE rolled loop:
// 4x ILP against the quarter-rate v_rcp chain, one loop-control overhead.
// Wave-uniform early exit: break when every lane's 4 chains have |del-1|<1e-6
// (single wave32 ballot, no divergence, deterministic); 16-step cap retained.
__device__ void betainc4(float a, float b, float lbeta,
                         const float edge[NEDGE], float cdf[NEDGE]) {
    const float FPMIN = 1.0e-30f;
    const float qab   = a + b;                      // swap-invariant
    const float thresh = (a + 1.0f) * frcp(qab + 2.0f);

    float xa[NEDGE], xb[NEDGE], xx[NEDGE], bt[NEDGE];
    bool  sw[NEDGE];
#pragma unroll
    for (int j = 0; j < NEDGE; ++j) {
        float x = edge[j];
        bool  s = x >= thresh;
        sw[j] = s;
        xa[j] = s ? b : a;
        xb[j] = s ? a : b;
        xx[j] = s ? (1.0f - x) : x;
        // bt identical for both orientations: a ln x + b ln(1-x) - ln B(a,b)
        float lbt = a * __logf(x) + b * __logf(1.0f - x) - lbeta;
        bt[j] = __expf(lbt);
    }

    float c[NEDGE], d[NEDGE], h[NEDGE], qap[NEDGE], qam[NEDGE];
#pragma unroll
    for (int j = 0; j < NEDGE; ++j) {
        qap[j] = xa[j] + 1.0f;
        qam[j] = xa[j] - 1.0f;
        c[j] = 1.0f;
        float dd = 1.0f - qab * xx[j] * frcp(qap[j]);
        if (fabsf(dd) < FPMIN) dd = FPMIN;
        d[j] = frcp(dd);
        h[j] = d[j];
    }

#pragma unroll 1
    for (int m = 1; m <= 16; ++m) {
        float fm = (float)m;
        float m2 = 2.0f * fm;
        float werr = 0.0f;
#pragma unroll
        for (int j = 0; j < NEDGE; ++j) {
            float aa = fm * (xb[j] - fm) * xx[j]
                     * frcp((qam[j] + m2) * (xa[j] + m2));
            float dd = 1.0f + aa * d[j];
            if (fabsf(dd) < FPMIN) dd = FPMIN;
            float cc = 1.0f + aa * frcp(c[j]);
            if (fabsf(cc) < FPMIN) cc = FPMIN;
            dd = frcp(dd);
            h[j] *= dd * cc;

            aa = -(xa[j] + fm) * (qab + fm) * xx[j]
               * frcp((xa[j] + m2) * (qap[j] + m2));
            float dd2 = 1.0f + aa * dd;
            if (fabsf(dd2) < FPMIN) dd2 = FPMIN;
            float cc2 = 1.0f + aa * frcp(cc);
            if (fabsf(cc2) < FPMIN) cc2 = FPMIN;
            dd2 = frcp(dd2);
            float del = dd2 * cc2;
            h[j] *= del;
            d[j] = dd2;
            c[j] = cc2;
            werr = fmaxf(werr, fabsf(del - 1.0f));
        }
        // Uniform break: all 32 lanes x 4 chains converged -> stop early.
        if (__all(werr < 1.0e-6f)) break;
    }

#pragma unroll
    for (int j = 0; j < NEDGE; ++j) {
        float r = bt[j] * h[j] * frcp(xa[j]);
        r = sw[j] ? (1.0f - r) : r;
        float x = edge[j];
        r = (x <= 0.0f) ? 0.0f : r;
        r = (x >= 1.0f) ? 1.0f : r;
        cdf[j] = r;
    }
}

// Compute one row; results land in mass[0..4], edge[0..4] (register arrays).
__device__ __forceinline__ void compute_sample(
        int u, int v, float a, float b,
        const float* __restrict__ uid_w, const float* __restrict__ iid_w,
        float mass[NBINS], float edge[NBINS]) {
    // Embedding rows: temporal loads — both tables (40 MB) stay L2-resident
    const float* pu = uid_w + (long long)u * NBINS;
    const float* pi = iid_w + (long long)v * NBINS;
    float l[NBINS];
#pragma unroll
    for (int j = 0; j < NBINS; ++j) l[j] = pu[j] + pi[j];

    // softmax (max-shifted) + cumsum -> edges
    float mx = l[0];
#pragma unroll
    for (int j = 1; j < NBINS; ++j) mx = fmaxf(mx, l[j]);
    float e[NBINS], ssum = 0.0f;
#pragma unroll
    for (int j = 0; j < NBINS; ++j) { e[j] = __expf(l[j] - mx); ssum += e[j]; }
    float inv = frcp(ssum);
    float acc = 0.0f;
#pragma unroll
    for (int j = 0; j < NBINS; ++j) { acc += e[j] * inv; edge[j] = acc; }

    // ln B(a,b) once per row
    float lbeta = lgamma_pos(a) + lgamma_pos(b) - lgamma_pos(a + b);

    float cdf[NEDGE];
    betainc4(a, b, lbeta, edge, cdf);

    mass[0] = cdf[0];
#pragma unroll
    for (int j = 1; j < NEDGE; ++j) mass[j] = cdf[j] - cdf[j - 1];
    mass[NEDGE] = 1.0f - cdf[NEDGE - 1];
}

__global__ __launch_bounds__(256) void BetaBinsMassAdaptive_kernel(
        const int* __restrict__ uid, const int* __restrict__ iid,
        const float* __restrict__ alpha, const float* __restrict__ beta,
        const float* __restrict__ uid_w, const float* __restrict__ iid_w,
        float* __restrict__ mass_out, float* __restrict__ edges_out, int B) {
    const int tid = blockIdx.x * blockDim.x + threadIdx.x;
    const int nthreads = gridDim.x * blockDim.x;
    const int npairs = B >> 1;

    for (int p = tid; p < npairs; p += nthreads) {
        // Prefetch the NEXT grid-stride iteration's random-gather rows
        // (global_prefetch_b8); ~2K cycles of CF compute hide the latency.
        int pn = p + nthreads;
        if (pn < npairs) {
            v2i un = __builtin_nontemporal_load((const v2i*)uid + pn);
            v2i vn = __builtin_nontemporal_load((const v2i*)iid + pn);
            __builtin_prefetch(uid_w + (long long)un.x * NBINS, 0, 3);
            __builtin_prefetch(uid_w + (long long)un.y * NBINS, 0, 3);
            __builtin_prefetch(iid_w + (long long)vn.x * NBINS, 0, 3);
            __builtin_prefetch(iid_w + (long long)vn.y * NBINS, 0, 3);
        }

        // Current pair: coalesced 8B non-temporal stream loads
        v2i u2 = __builtin_nontemporal_load((const v2i*)uid + p);
        v2i v2 = __builtin_nontemporal_load((const v2i*)iid + p);
        v2f a2 = __builtin_nontemporal_load((const v2f*)alpha + p);
        v2f b2 = __builtin_nontemporal_load((const v2f*)beta + p);

        float m01[2 * NBINS], e01[2 * NBINS];
        compute_sample(u2.x, v2.x, a2.x, b2.x, uid_w, iid_w, m01,         e01);
        compute_sample(u2.y, v2.y, a2.y, b2.y, uid_w, iid_w, m01 + NBINS, e01 + NBINS);

        // Two adjacent rows = 40 contiguous, 8B-aligned bytes per output:
        // 5x b64 non-temporal stores each (don't pollute L2 tables).
        v2f* mrow = (v2f*)(mass_out  + (size_t)p * (2 * NBINS));
        v2f* erow = (v2f*)(edges_out + (size_t)p * (2 * NBINS));
#pragma unroll
        for (int k = 0; k < NBINS; ++k) {
            v2f mv; mv.x = m01[2 * k]; mv.y = m01[2 * k + 1];
            v2f ev; ev.x = e01[2 * k]; ev.y = e01[2 * k + 1];
            __builtin_nontemporal_store(mv, mrow + k);
            __builtin_nontemporal_store(ev, erow + k);
        }
    }

    // Odd-B tail (not hit for B = 4,194,304, kept for robustness)
    if ((B & 1) && tid == 0) {
        int s = B - 1;
        float m[NBINS], e[NBINS];
        compute_sample(uid[s], iid[s], alpha[s], beta[s], uid_w, iid_w, m, e);
#pragma unroll
        for (int j = 0; j < NBINS; ++j) {
            __builtin_nontemporal_store(m[j], mass_out  + (size_t)s * NBINS + j);
            __builtin_nontemporal_store(e[j], edges_out + (size_t)s * NBINS + j);
        }
    }
}

extern "C" void kernel_launch(void* const* d_in, const int* in_sizes, int n_in,
                              void* d_out, int out_size, void* d_ws, size_t ws_size,
                              hipStream_t stream) {
    const int*   uid   = (const int*)d_in[0];
    const int*   iid   = (const int*)d_in[1];
    const float* alpha = (const float*)d_in[2];
    const float* beta  = (const float*)d_in[3];
    const float* uid_w = (const float*)d_in[4];
    const float* iid_w = (const float*)d_in[5];

    const int B = in_sizes[0];
    float* out   = (float*)d_out;
    float* mass  = out;                          // output 0: mass  [B,5]
    float* edges = out + (size_t)B * NBINS;      // output 1: edges [B,5]

    const int threads = 256;                     // 8 wave32 waves per block
    int npairs = B >> 1;
    int blocks = 2048;                           // grid-stride; 16K waves in flight
    int need = (npairs + threads - 1) / threads;
    if (need > 0 && need < blocks) blocks = need;
    if (blocks < 1) blocks = 1;

    BetaBinsMassAdaptive_kernel<<<blocks, threads, 0, stream>>>(
        uid, iid, alpha, beta, uid_w, iid_w, mass, edges, B);
}